// EarthMoverDistance_63617055588919
// MI455X (gfx1250) — compile-verified
//
#include <hip/hip_runtime.h>
#include <math.h>

typedef __attribute__((ext_vector_type(2))) float v2f;
typedef __attribute__((ext_vector_type(8))) float v8f;

#define WAVES 4  // waves (wave32) per block

__device__ __forceinline__ float red16(float x) {
  // sum across the 16 lanes of each half-wave (masks < 16 keep lane bit 4)
  x += __shfl_xor(x, 1, 32);
  x += __shfl_xor(x, 2, 32);
  x += __shfl_xor(x, 4, 32);
  x += __shfl_xor(x, 8, 32);
  return x;
}

// Pack: X4[i] = (x, y, z, |x|^2) ; Y4[j] = (-2yx, -2yy, -2yz, |y|^2)
__global__ void emd_pack(const float* __restrict__ in1, const float* __restrict__ in2,
                         float* __restrict__ X4, float* __restrict__ Y4,
                         int bn, int bm) {
  int i = blockIdx.x * blockDim.x + threadIdx.x;
  if (i < bn) {
    float x = in1[3 * i + 0], y = in1[3 * i + 1], z = in1[3 * i + 2];
    X4[4 * i + 0] = x; X4[4 * i + 1] = y; X4[4 * i + 2] = z;
    X4[4 * i + 3] = x * x + y * y + z * z;
  }
  if (i < bm) {
    float x = in2[3 * i + 0], y = in2[3 * i + 1], z = in2[3 * i + 2];
    Y4[4 * i + 0] = -2.f * x; Y4[4 * i + 1] = -2.f * y; Y4[4 * i + 2] = -2.f * z;
    Y4[4 * i + 3] = x * x + y * y + z * z;
  }
}

__global__ void emd_init(float* __restrict__ satl, float* __restrict__ satr,
                         float* __restrict__ wsCost, int bn, int bm, int strips,
                         float fl, float fr) {
  int i = blockIdx.x * blockDim.x + threadIdx.x;
  if (i < bn) satl[i] = fl;
  if (i < bm) satr[i] = fr;
  if (i < strips) wsCost[i] = 0.f;
}

// Per level: Y4L[j] = (lv2 * -2y, lv2*|y|^2 + log2(satr[j]))  -> P1's B fragment
__global__ void emd_ypack(const float* __restrict__ Y4, const float* __restrict__ satr,
                          float* __restrict__ Y4L, float lv2, int bm) {
  int j = blockIdx.x * blockDim.x + threadIdx.x;
  if (j < bm) {
    Y4L[4 * j + 0] = lv2 * Y4[4 * j + 0];
    Y4L[4 * j + 1] = lv2 * Y4[4 * j + 1];
    Y4L[4 * j + 2] = lv2 * Y4[4 * j + 2];
    Y4L[4 * j + 3] = lv2 * Y4[4 * j + 3] + __builtin_amdgcn_logf(satr[j]);
  }
}

// Pass 1: rowsum[i] = sum_j exp2( lv2*d2 + log2 satr[j] )
// WMMA emits the exp2 argument directly: A=[x,y,z,1], B=Y4L, C=lv2*|x|^2
__global__ void __launch_bounds__(WAVES * 32)
emd_p1(const float* __restrict__ X4, const float* __restrict__ Y4L,
       float* __restrict__ rowsum, float lv2, int n, int m, int totalStrips) {
  const int lane = threadIdx.x & 31;
  const int strip = blockIdx.x * WAVES + (threadIdx.x >> 5);
  if (strip >= totalStrips) return;  // wave-uniform
  const int spb = n >> 4;
  const int batch = strip / spb;
  const int r0 = (strip % spb) << 4;
  const int lc = lane & 15, hi = lane >> 4;

  const float* X4b = X4 + (size_t)batch * n * 4;
  const float* Y4Lb = Y4L + (size_t)batch * m * 4;

  v2f a = *(const v2f*)(X4b + (size_t)(r0 + lc) * 4 + 2 * hi);
  a.y = hi ? 1.0f : a.y;
  v8f c;
#pragma unroll
  for (int v = 0; v < 8; ++v)
    c[v] = lv2 * X4b[(size_t)(r0 + v + 8 * hi) * 4 + 3];

  float acc[8];
#pragma unroll
  for (int v = 0; v < 8; ++v) acc[v] = 0.f;

  for (int t = 0; t < (m >> 4); ++t) {
    const int col = (t << 4) + lc;
    v2f bv = *(const v2f*)(Y4Lb + (size_t)col * 4 + 2 * hi);
    v8f d = __builtin_amdgcn_wmma_f32_16x16x4_f32(false, a, false, bv, (short)0, c,
                                                  false, false);
#pragma unroll
    for (int v = 0; v < 8; ++v) acc[v] += __builtin_amdgcn_exp2f(d[v]);
  }
#pragma unroll
  for (int v = 0; v < 8; ++v) acc[v] = red16(acc[v]);
  if (lc == 0) {
    float* rb = rowsum + (size_t)batch * n;
#pragma unroll
    for (int v = 0; v < 8; ++v) rb[r0 + v + 8 * hi] = acc[v];
  }
}

// frow = satl/(rowsum+1e-9); X4L[i] = (x,y,z, lv2*|x|^2 + log2 frow[i]) -> P2's A
__global__ void emd_frow(const float* __restrict__ satl, const float* __restrict__ rowsum,
                         const float* __restrict__ X4, float* __restrict__ frow,
                         float* __restrict__ X4L, float lv2, int bn) {
  int i = blockIdx.x * blockDim.x + threadIdx.x;
  if (i < bn) {
    float fr = satl[i] / (rowsum[i] + 1e-9f);
    frow[i] = fr;
    X4L[4 * i + 0] = X4[4 * i + 0];
    X4L[4 * i + 1] = X4[4 * i + 1];
    X4L[4 * i + 2] = X4[4 * i + 2];
    X4L[4 * i + 3] = lv2 * X4[4 * i + 3] + __builtin_amdgcn_logf(fr);
  }
}

// Pass 2: colraw[j] = sum_i exp2( lv2*d2 + log2 frow[i] )  (column strips)
// A=X4L (per tile), B'=[lv2*-2y, 1] (per wave), C'=lv2*|y|^2 (per wave)
__global__ void __launch_bounds__(WAVES * 32)
emd_p2(const float* __restrict__ X4L, const float* __restrict__ Y4,
       float* __restrict__ colraw, float lv2, int n, int m, int totalStrips) {
  const int lane = threadIdx.x & 31;
  const int strip = blockIdx.x * WAVES + (threadIdx.x >> 5);
  if (strip >= totalStrips) return;
  const int spb = m >> 4;
  const int batch = strip / spb;
  const int c0 = (strip % spb) << 4;
  const int lc = lane & 15, hi = lane >> 4;

  const float* X4Lb = X4L + (size_t)batch * n * 4;
  const float* Y4b = Y4 + (size_t)batch * m * 4;

  const int col = c0 + lc;
  v2f bv = *(const v2f*)(Y4b + (size_t)col * 4 + 2 * hi);
  bv.x = lv2 * bv.x;
  bv.y = hi ? 1.0f : lv2 * bv.y;
  float yn = lv2 * Y4b[(size_t)col * 4 + 3];
  v8f c;
#pragma unroll
  for (int v = 0; v < 8; ++v) c[v] = yn;

  float acc[8];
#pragma unroll
  for (int v = 0; v < 8; ++v) acc[v] = 0.f;

  for (int t = 0; t < (n >> 4); ++t) {
    const int r0 = t << 4;
    v2f a = *(const v2f*)(X4Lb + (size_t)(r0 + lc) * 4 + 2 * hi);
    v8f d = __builtin_amdgcn_wmma_f32_16x16x4_f32(false, a, false, bv, (short)0, c,
                                                  false, false);
#pragma unroll
    for (int v = 0; v < 8; ++v) acc[v] += __builtin_amdgcn_exp2f(d[v]);
  }
  float s = 0.f;
#pragma unroll
  for (int v = 0; v < 8; ++v) s += acc[v];
  s += __shfl_xor(s, 16, 32);
  if (lane < 16) colraw[(size_t)batch * m + c0 + lane] = s;
}

// ss = satr*colraw + 1e-9 ; lsr2 = log2(satr^2/ss) ; satr -= sum_i w
__global__ void emd_sr2(float* __restrict__ satr, const float* __restrict__ colraw,
                        float* __restrict__ lsr2, int bm) {
  int j = blockIdx.x * blockDim.x + threadIdx.x;
  if (j < bm) {
    float s = satr[j], cr = colraw[j];
    float ss = s * cr + 1e-9f;
    float r2 = s * s / ss;
    lsr2[j] = __builtin_amdgcn_logf(r2);
    satr[j] = fmaxf(s - r2 * cr, 0.f);
  }
}

// Pass 3: raw-d2 WMMA; e = exp2(fma(lv2,d2,lsr2[j])); rowdep + cost; satl update
__global__ void __launch_bounds__(WAVES * 32)
emd_p3(const float* __restrict__ X4, const float* __restrict__ Y4,
       const float* __restrict__ lsr2, const float* __restrict__ frow,
       float* __restrict__ satl, float* __restrict__ wsCost,
       float lv2, int n, int m, int totalStrips) {
  const int lane = threadIdx.x & 31;
  const int strip = blockIdx.x * WAVES + (threadIdx.x >> 5);
  if (strip >= totalStrips) return;
  const int spb = n >> 4;
  const int batch = strip / spb;
  const int r0 = (strip % spb) << 4;
  const int lc = lane & 15, hi = lane >> 4;

  const float* X4b = X4 + (size_t)batch * n * 4;
  const float* Y4b = Y4 + (size_t)batch * m * 4;
  const float* lsr2b = lsr2 + (size_t)batch * m;

  v2f a = *(const v2f*)(X4b + (size_t)(r0 + lc) * 4 + 2 * hi);
  a.y = hi ? 1.0f : a.y;
  v8f c;
#pragma unroll
  for (int v = 0; v < 8; ++v) c[v] = X4b[(size_t)(r0 + v + 8 * hi) * 4 + 3];

  float acc[8], cacc[8];
#pragma unroll
  for (int v = 0; v < 8; ++v) { acc[v] = 0.f; cacc[v] = 0.f; }

  for (int t = 0; t < (m >> 4); ++t) {
    const int col = (t << 4) + lc;
    v2f bv = *(const v2f*)(Y4b + (size_t)col * 4 + 2 * hi);
    float l2 = lsr2b[col];
    v8f d = __builtin_amdgcn_wmma_f32_16x16x4_f32(false, a, false, bv, (short)0, c,
                                                  false, false);
#pragma unroll
    for (int v = 0; v < 8; ++v) {
      float e = __builtin_amdgcn_exp2f(fmaf(lv2, d[v], l2));
      acc[v] += e;
      cacc[v] += e * sqrtf(fmaxf(d[v], 1e-12f));
    }
  }
#pragma unroll
  for (int v = 0; v < 8; ++v) { acc[v] = red16(acc[v]); cacc[v] = red16(cacc[v]); }

  float cpart = 0.f;
  if (lc == 0) {
    const float* frowb = frow + (size_t)batch * n;
    float* satlb = satl + (size_t)batch * n;
#pragma unroll
    for (int v = 0; v < 8; ++v) {
      int row = r0 + v + 8 * hi;
      float fr = frowb[row];
      satlb[row] = fmaxf(satlb[row] - fr * acc[v], 0.f);
      cpart += fr * cacc[v];
    }
  }
  cpart += __shfl_xor(cpart, 16, 32);  // lane0 <- lane0 + lane16
  if (lane == 0) wsCost[strip] += cpart;  // single writer; deterministic across levels
}

__global__ void emd_reduce(const float* __restrict__ wsCost, float* __restrict__ out,
                           int spb) {
  __shared__ float sm[256];
  float s = 0.f;
  for (int i = threadIdx.x; i < spb; i += blockDim.x)
    s += wsCost[(size_t)blockIdx.x * spb + i];
  sm[threadIdx.x] = s;
  __syncthreads();
  for (int off = 128; off > 0; off >>= 1) {
    if (threadIdx.x < off) sm[threadIdx.x] += sm[threadIdx.x + off];
    __syncthreads();
  }
  if (threadIdx.x == 0) out[blockIdx.x] = sm[0];
}

extern "C" void kernel_launch(void* const* d_in, const int* in_sizes, int n_in,
                              void* d_out, int out_size, void* d_ws, size_t ws_size,
                              hipStream_t stream) {
  const float* in1 = (const float*)d_in[0];
  const float* in2 = (const float*)d_in[1];
  const int b = out_size;  // [b] output
  const int n = in_sizes[0] / (3 * b);
  const int m = in_sizes[1] / (3 * b);
  const int bn = b * n, bm = b * m;
  const int mx = (n > m) ? n : m;
  const float fl = (float)(mx / n), fr = (float)(mx / m);

  float* ws = (float*)d_ws;
  float* X4 = ws;     ws += (size_t)bn * 4;
  float* Y4 = ws;     ws += (size_t)bm * 4;
  float* X4L = ws;    ws += (size_t)bn * 4;
  float* Y4L = ws;    ws += (size_t)bm * 4;
  float* satl = ws;   ws += bn;
  float* satr = ws;   ws += bm;
  float* rowsum = ws; ws += bn;
  float* frow = ws;   ws += bn;
  float* colraw = ws; ws += bm;
  float* lsr2 = ws;   ws += bm;
  const int rStrips = bn >> 4;  // b*(n/16)
  const int cStrips = bm >> 4;
  float* wsCost = ws; ws += rStrips;

  int packN = (bn > bm) ? bn : bm;
  emd_pack<<<(packN + 255) / 256, 256, 0, stream>>>(in1, in2, X4, Y4, bn, bm);
  int initN = packN > rStrips ? packN : rStrips;
  emd_init<<<(initN + 255) / 256, 256, 0, stream>>>(satl, satr, wsCost, bn, bm,
                                                    rStrips, fl, fr);

  float levels[11];
  int k = 0;
  for (int j = 8; j >= -1; --j) levels[k++] = -powf(4.f, (float)j);
  levels[k++] = 0.f;

  const float LOG2E = 1.44269504088896340736f;
  const int rBlocks = (rStrips + WAVES - 1) / WAVES;
  const int cBlocks = (cStrips + WAVES - 1) / WAVES;
  for (int L = 0; L < 11; ++L) {
    const float lv2 = levels[L] * LOG2E;
    emd_ypack<<<(bm + 255) / 256, 256, 0, stream>>>(Y4, satr, Y4L, lv2, bm);
    emd_p1<<<rBlocks, WAVES * 32, 0, stream>>>(X4, Y4L, rowsum, lv2, n, m, rStrips);
    emd_frow<<<(bn + 255) / 256, 256, 0, stream>>>(satl, rowsum, X4, frow, X4L, lv2,
                                                   bn);
    emd_p2<<<cBlocks, WAVES * 32, 0, stream>>>(X4L, Y4, colraw, lv2, n, m, cStrips);
    emd_sr2<<<(bm + 255) / 256, 256, 0, stream>>>(satr, colraw, lsr2, bm);
    emd_p3<<<rBlocks, WAVES * 32, 0, stream>>>(X4, Y4, lsr2, frow, satl, wsCost, lv2,
                                               n, m, rStrips);
  }
  emd_reduce<<<b, 256, 0, stream>>>(wsCost, (float*)d_out, n >> 4);
}